// ROIPoolingLayer_80848464380136
// MI455X (gfx1250) — compile-verified
//
#include <hip/hip_runtime.h>
#include <stdint.h>

// ROI pooling (nearest-neighbor sampling), NHWC.
//   fm:   (N, 64, 64, 256) f32
//   rois: (N, 4)           f32  [x1, y1, x2, y2]
//   out:  (N, 7, 7, 256)   f32
// Pure gather-copy: each (n,i,j) bin copies one contiguous 1KB channel vector.
// ~25.7 MB total traffic -> HBM-bound (~1.1us at 23.3 TB/s). No matmul -> no
// WMMA; the CDNA5-specific win is the async global->LDS->global DMA path
// (ASYNCcnt-tracked), which moves data without a VGPR round trip.

#define PH 7
#define PW 7
#define HH 64
#define WW 64
#define CC 256

#if defined(__has_builtin)
#  if __has_builtin(__builtin_amdgcn_global_load_async_to_lds_b128) && \
      __has_builtin(__builtin_amdgcn_global_store_async_from_lds_b128) && \
      __has_builtin(__builtin_amdgcn_s_wait_asynccnt)
#    define USE_ASYNC_LDS 1
#  endif
#endif

// The async builtins take 128-bit payload-typed pointers (int4*), per the
// round-1 diagnostic: "cannot initialize a parameter of type
// 'vector_size(4*sizeof(int)) int *' with ... 'void *'".
typedef int v4i __attribute__((ext_vector_type(4)));
typedef __attribute__((address_space(1))) v4i gv4i;  // global
typedef __attribute__((address_space(3))) v4i lv4i;  // LDS

__global__ __launch_bounds__(256) void roi_pool_async_kernel(
    const float* __restrict__ fm, const float* __restrict__ rois,
    float* __restrict__ out, int nBins) {
  // 8 waves per block, one 1KB LDS slice per wave.
  __shared__ float tile[8][CC];

  const int wave = threadIdx.x >> 5;   // wave32
  const int lane = threadIdx.x & 31;
  const int bin  = blockIdx.x * 8 + wave;
  if (bin >= nBins) return;            // uniform per-wave branch

  const int n  = bin / (PH * PW);
  const int ij = bin - n * (PH * PW);
  const int i  = ij / PW;
  const int j  = ij - i * PW;

  // All lanes of the wave compute the same indices (no divergence).
  const float x1 = rois[n * 4 + 0];
  const float y1 = rois[n * 4 + 1];
  const float x2 = rois[n * 4 + 2];
  const float y2 = rois[n * 4 + 3];

  // t = i/7 in f32, round-half-to-even (v_rndne) matches jnp.round, then clip.
  int gh = (int)rintf(y1 + (y2 - y1) * ((float)i / (float)PH));
  int gw = (int)rintf(x1 + (x2 - x1) * ((float)j / (float)PW));
  gh = gh < 0 ? 0 : (gh > HH - 1 ? HH - 1 : gh);
  gw = gw < 0 ? 0 : (gw > WW - 1 ? WW - 1 : gw);

  // 1KB contiguous segment: lane handles 16B at lane*16 and lane*16+512.
  const float* src = fm + (((size_t)n * HH + gh) * WW + gw) * CC + lane * 4;
  float*       dst = out + (size_t)bin * CC + lane * 4;

#if defined(USE_ASYNC_LDS)
  // Flat LDS address carries the LDS byte offset in bits [31:0] (ISA 10.2);
  // AS3 pointers are 32-bit, so the uintptr_t round trip yields the offset.
  lv4i* l  = (lv4i*)(uintptr_t)&tile[wave][lane * 4];
  gv4i* gs = (gv4i*)(uintptr_t)src;
  gv4i* gd = (gv4i*)(uintptr_t)dst;
  // INST_OFFSET is added to BOTH the global and LDS addresses (ISA 08 §4.4).
  __builtin_amdgcn_global_load_async_to_lds_b128(gs, l, 0, 0);
  __builtin_amdgcn_global_load_async_to_lds_b128(gs, l, 512, 0);
  __builtin_amdgcn_s_wait_asynccnt(0);  // loads landed in our private LDS slice
  __builtin_amdgcn_global_store_async_from_lds_b128(gd, l, 0, 0);
  __builtin_amdgcn_global_store_async_from_lds_b128(gd, l, 512, 0);
  // S_ENDPGM's implicit wait-idle drains ASYNCcnt before wave termination.
#else
  // Fallback: direct 128-bit VGPR copy (still bandwidth-optimal).
  float4 a = *(const float4*)src;
  float4 b = *(const float4*)(src + 128);
  *(float4*)dst = a;
  *(float4*)(dst + 128) = b;
  (void)tile;
#endif
}

extern "C" void kernel_launch(void* const* d_in, const int* in_sizes, int n_in,
                              void* d_out, int out_size, void* d_ws, size_t ws_size,
                              hipStream_t stream) {
  const float* fm   = (const float*)d_in[0];
  const float* rois = (const float*)d_in[1];
  float* out = (float*)d_out;

  const int N     = in_sizes[1] / 4;      // rois is (N,4)
  const int nBins = N * PH * PW;          // one wave per bin
  const int blocks = (nBins + 7) / 8;     // 8 waves (256 threads) per block

  roi_pool_async_kernel<<<blocks, 256, 0, stream>>>(fm, rois, out, nBins);
  (void)d_ws; (void)ws_size; (void)n_in; (void)out_size;
}